// CollectDiffuseAttention_38233798869797
// MI455X (gfx1250) — compile-verified
//
#include <hip/hip_runtime.h>

// CollectDiffuseAttention: B=128, L=4096, C=384, single query per batch.
// out[b,c] = sum_l softmax_l(q[b]·kc[b,l]/T) * v[b,l,c]
// Memory-bound (AI ~0.5 FLOP/B, 1.61 GB streamed, ~69 us floor @ 23.3 TB/s):
// flash-decoding split over L, one streaming pass over K and V,
// WMMA f32 16x16x4 for the q·K dot products, non-temporal K/V loads
// (read-once stream >> 192 MB L2).

typedef __attribute__((ext_vector_type(2))) float v2f;
typedef __attribute__((ext_vector_type(4))) float v4f;
typedef __attribute__((ext_vector_type(8))) float v8f;

#define B_DIM 128
#define L_DIM 4096
#define C_DIM 384
#define TEMP_INV 0.05103103630798288f  // 1/sqrt(384)

#define WAVES 8
#define THREADS (WAVES * 32)
#define TILE_M 16
#define PART_STRIDE (C_DIM + 2)  // [m, s, acc[384]] per (batch, slice)

__global__ __launch_bounds__(THREADS)
void cda_partial_kernel(const float* __restrict__ q,
                        const float* __restrict__ kc,
                        const float* __restrict__ v,
                        float* __restrict__ ws,
                        int slices, int rows_per_slice) {
  __shared__ float sq[C_DIM];
  __shared__ float red_m[WAVES];
  __shared__ float red_s[WAVES];
  __shared__ float red_acc[WAVES][C_DIM];

  const int b    = blockIdx.y;
  const int sl   = blockIdx.x;
  const int tid  = threadIdx.x;
  const int wave = tid >> 5;
  const int lane = tid & 31;
  const int half = lane >> 4;   // 0: lanes 0-15, 1: lanes 16-31
  const int mrow = lane & 15;   // row within 16-row tile (A-matrix M index)

  // Stage q (pre-scaled by 1/temperature) into LDS for B-fragment broadcast.
  for (int c = tid; c < C_DIM; c += THREADS)
    sq[c] = q[(size_t)b * C_DIM + c] * TEMP_INV;
  __syncthreads();

  const int row0  = sl * rows_per_slice;
  const int tiles = rows_per_slice / TILE_M;

  float m = -__builtin_inff();
  float s = 0.f;
  float acc[12];
#pragma unroll
  for (int j = 0; j < 12; ++j) acc[j] = 0.f;

  for (int t = wave; t < tiles; t += WAVES) {
    const int r0 = row0 + t * TILE_M;

    // ---- logits for 16 rows via WMMA f32 16x16x4, accumulated over C ----
    // A layout (32-bit A 16x4): lane m (0-15) holds K=0,1 ; lane m+16 holds K=2,3
    // Lane pair (m, m+16) covers 128 contiguous bytes per 8-chunk clause.
    const float* krow =
        kc + ((size_t)b * L_DIM + (size_t)(r0 + mrow)) * C_DIM + 2 * half;
    v8f d = {};
#pragma unroll 8
    for (int kk = 0; kk < C_DIM / 4; ++kk) {
      // Non-temporal: K is a read-once stream, keep it out of L2/WGP$.
      v2f a = __builtin_nontemporal_load((const v2f*)(krow + kk * 4));
      v2f bf;                            // q broadcast across all 16 columns
      bf.x = sq[kk * 4 + 2 * half + 0];  // -> ds_load_2addr_b64 broadcast
      bf.y = sq[kk * 4 + 2 * half + 1];
      d = __builtin_amdgcn_wmma_f32_16x16x4_f32(
          false, a, false, bf, (short)0, d, false, false);
    }

    // D: vgpr r, lanes 0-15 = row r ; lanes 16-31 = row 8+r (columns identical).
    // Broadcast all 16 logits to every lane with one half-swap per register.
    float lg[16];
#pragma unroll
    for (int r = 0; r < 8; ++r) {
      float own = d[r];
      float oth = __shfl_xor(own, 16, 32);
      lg[r]     = (half == 0) ? own : oth;
      lg[8 + r] = (half == 0) ? oth : own;
    }

    // ---- online softmax update (wave-uniform m, s) ----
    float tmax = lg[0];
#pragma unroll
    for (int i = 1; i < 16; ++i) tmax = fmaxf(tmax, lg[i]);
    const float mnew  = fmaxf(m, tmax);
    const float scale = __expf(m - mnew);  // 0 on first tile (m = -inf)
    float w[16];
    float wsum = 0.f;
#pragma unroll
    for (int i = 0; i < 16; ++i) {
      w[i] = __expf(lg[i] - mnew);
      wsum += w[i];
    }
    s = s * scale + wsum;
    m = mnew;
#pragma unroll
    for (int j = 0; j < 12; ++j) acc[j] *= scale;

    // ---- accumulate attn * V : lane owns channels [lane*12, lane*12+12) ----
    const float* vbase =
        v + ((size_t)b * L_DIM + (size_t)r0) * C_DIM + lane * 12;
#pragma unroll 4
    for (int i = 0; i < 16; ++i) {
      const v4f* vp = (const v4f*)(vbase + (size_t)i * C_DIM);  // 16B aligned
      // Non-temporal b128 loads: V is a read-once stream.
      v4f p0 = __builtin_nontemporal_load(vp + 0);
      v4f p1 = __builtin_nontemporal_load(vp + 1);
      v4f p2 = __builtin_nontemporal_load(vp + 2);
      const float wi = w[i];
#pragma unroll
      for (int e = 0; e < 4; ++e) {
        acc[e + 0] = fmaf(wi, p0[e], acc[e + 0]);
        acc[e + 4] = fmaf(wi, p1[e], acc[e + 4]);
        acc[e + 8] = fmaf(wi, p2[e], acc[e + 8]);
      }
    }
  }

  // ---- combine the 8 waves' online-softmax states within the block ----
  if (lane == 0) { red_m[wave] = m; red_s[wave] = s; }
#pragma unroll
  for (int j = 0; j < 12; ++j) red_acc[wave][lane * 12 + j] = acc[j];
  __syncthreads();

  float M = red_m[0];
#pragma unroll
  for (int wv = 1; wv < WAVES; ++wv) M = fmaxf(M, red_m[wv]);
  float ef[WAVES];
  float S = 0.f;
#pragma unroll
  for (int wv = 0; wv < WAVES; ++wv) {
    ef[wv] = __expf(red_m[wv] - M);
    S += ef[wv] * red_s[wv];
  }

  float* part = ws + ((size_t)b * slices + sl) * PART_STRIDE;
  if (tid == 0) { part[0] = M; part[1] = S; }
  for (int c = tid; c < C_DIM; c += THREADS) {
    float a = 0.f;
#pragma unroll
    for (int wv = 0; wv < WAVES; ++wv) a += ef[wv] * red_acc[wv][c];
    part[2 + c] = a;
  }
}

// Fold the per-slice partials: out[b,c] = sum_sl e^{m_sl-M} acc_sl[c] / sum_sl e^{m_sl-M} s_sl
__global__ __launch_bounds__(C_DIM)
void cda_combine_kernel(const float* __restrict__ ws, float* __restrict__ out,
                        int slices) {
  const int b = blockIdx.x;
  const int c = threadIdx.x;
  const float* p = ws + (size_t)b * slices * PART_STRIDE;

  float M = -__builtin_inff();
  for (int sl = 0; sl < slices; ++sl)
    M = fmaxf(M, p[(size_t)sl * PART_STRIDE]);
  float S = 0.f, a = 0.f;
  for (int sl = 0; sl < slices; ++sl) {
    const float e = __expf(p[(size_t)sl * PART_STRIDE] - M);
    S += e * p[(size_t)sl * PART_STRIDE + 1];
    a += e * p[(size_t)sl * PART_STRIDE + 2 + c];
  }
  out[(size_t)b * C_DIM + c] = a / S;
}

extern "C" void kernel_launch(void* const* d_in, const int* in_sizes, int n_in,
                              void* d_out, int out_size, void* d_ws,
                              size_t ws_size, hipStream_t stream) {
  (void)in_sizes; (void)n_in; (void)out_size;
  const float* q  = (const float*)d_in[0];
  const float* kc = (const float*)d_in[1];
  const float* v  = (const float*)d_in[2];
  float* out = (float*)d_out;
  float* ws  = (float*)d_ws;

  // Pick the largest slice count whose partials fit in the workspace.
  int slices = 16;
  while (slices > 1 &&
         (size_t)B_DIM * slices * PART_STRIDE * sizeof(float) > ws_size)
    slices >>= 1;
  const int rows_per_slice = L_DIM / slices;

  dim3 grid(slices, B_DIM);
  cda_partial_kernel<<<grid, THREADS, 0, stream>>>(q, kc, v, ws, slices,
                                                   rows_per_slice);
  cda_combine_kernel<<<B_DIM, C_DIM, 0, stream>>>(ws, out, slices);
}